// TCVAE_58626303590481
// MI455X (gfx1250) — compile-verified
//
#include <hip/hip_runtime.h>
#include <hip/hip_bf16.h>
#include <math.h>
#include <stdint.h>

typedef __attribute__((ext_vector_type(2))) float v2f;
typedef __attribute__((ext_vector_type(8))) float v8f;
typedef __attribute__((ext_vector_type(4))) int   v4i;

#define KDIM 32
#define FDIM 64
#define TJ   64
#define NEG_HALF_LOG_2PI (-0.91893853320467274178f)

#if defined(__HIP_DEVICE_COMPILE__) && __has_builtin(__builtin_amdgcn_global_load_async_to_lds_b128)
#define HAVE_ASYNC_LDS 1
#else
#define HAVE_ASYNC_LDS 0
#endif

__device__ __forceinline__ float wave_sum(float v) {
#pragma unroll
    for (int o = 16; o > 0; o >>= 1) v += __shfl_down(v, o, 32);
    return v;
}

#if HAVE_ASYNC_LDS
typedef __attribute__((address_space(1))) v4i* gv4i_p;
typedef __attribute__((address_space(3))) v4i* lv4i_p;
__device__ __forceinline__ void async_ld_b128(const float* g, float* lds) {
    __builtin_amdgcn_global_load_async_to_lds_b128((gv4i_p)g, (lv4i_p)lds, 0, 0);
}
__device__ __forceinline__ void wait_async0() {
#if __has_builtin(__builtin_amdgcn_s_wait_asynccnt)
    __builtin_amdgcn_s_wait_asynccnt(0);
#else
    asm volatile("s_wait_asynccnt 0x0" ::: "memory");
#endif
}
#endif

// ---------------------------------------------------------------------------
// Kernel 1: per-sample factors.
//   sigma = softplus(rho); z = mu + sigma*eps
//   A = -0.5/sigma^2 ; B = mu/sigma^2 ; C = -0.5*log2pi - log(sigma) - 0.5*mu^2/sigma^2
//   X[i,f]  = [z^2 | z]                       (row-major, WMMA A operand source)
//   Wp[f/2][j][2] = interleaved [A ; B] pairs (WMMA B operand: one b64/lane/k-step)
//   Aj/Bj/Cj = j-major copies for the streaming per-k LSE kernel
// ---------------------------------------------------------------------------
__global__ void k_prep(const float* __restrict__ mu_q, const float* __restrict__ rho_q,
                       const float* __restrict__ eps,
                       float* __restrict__ X, float* __restrict__ Wp,
                       float* __restrict__ Aj, float* __restrict__ Bj, float* __restrict__ Cj,
                       float* __restrict__ Csum, float* __restrict__ lqx, float* __restrict__ lpz,
                       int I) {
    int i = blockIdx.x;
    int k = threadIdx.x;               // wave32: lane == k
    float mu  = mu_q[i * KDIM + k];
    float rho = rho_q[i * KDIM + k];
    float ep  = eps[i * KDIM + k];
    float sig = (rho > 0.f) ? (rho + log1pf(__expf(-rho))) : log1pf(__expf(rho));
    float z   = fmaf(sig, ep, mu);
    float z2  = z * z;
    float inv = 1.f / (sig * sig);
    float Af  = -0.5f * inv;
    float Bf  = mu * inv;
    float logs = __logf(sig);
    float Cf  = NEG_HALF_LOG_2PI - logs - 0.5f * mu * mu * inv;

    X[i * FDIM + k]        = z2;
    X[i * FDIM + KDIM + k] = z;
    // feature f = k carries Af, feature f = 32+k carries Bf; pair = f>>1, slot = f&1
    Wp[(size_t)(k >> 1) * (2 * I) + 2 * i + (k & 1)]          = Af;
    Wp[(size_t)((KDIM + k) >> 1) * (2 * I) + 2 * i + (k & 1)] = Bf;
    Aj[i * KDIM + k] = Af;
    Bj[i * KDIM + k] = Bf;
    Cj[i * KDIM + k] = Cf;

    float cs = wave_sum(Cf);
    float qx = wave_sum(NEG_HALF_LOG_2PI - logs - 0.5f * ep * ep);
    float pz = wave_sum(NEG_HALF_LOG_2PI - 0.5f * z2);
    if (k == 0) { Csum[i] = cs; lqx[i] = qx; lpz[i] = pz; }
}

// ---------------------------------------------------------------------------
// Kernel 2: per-(i,k) online logsumexp over j of C+A*z2+B*z, summed over k.
// Wave per sample i, lane = k. 8 waves/block share j-tiles staged in LDS via
// GLOBAL_LOAD_ASYNC_TO_LDS_B128 (ASYNCcnt) when available.
// ---------------------------------------------------------------------------
__global__ void k_perk_lse(const float* __restrict__ X,
                           const float* __restrict__ Aj, const float* __restrict__ Bj,
                           const float* __restrict__ Cj,
                           float* __restrict__ lprod, float log_norm, int I) {
    __shared__ __align__(16) float sA[TJ * KDIM];
    __shared__ __align__(16) float sB[TJ * KDIM];
    __shared__ __align__(16) float sC[TJ * KDIM];
    int w    = threadIdx.x >> 5;
    int lane = threadIdx.x & 31;
    int i    = blockIdx.x * 8 + w;
    float z1 = X[i * FDIM + KDIM + lane];
    float z2 = X[i * FDIM + lane];
    float m = -INFINITY, s = 0.f;
    for (int j0 = 0; j0 < I; j0 += TJ) {
        __syncthreads();
#if HAVE_ASYNC_LDS
        {
            int t4 = threadIdx.x * 4;                  // 256 thr * 16B = 4KB/round
#pragma unroll
            for (int r = 0; r < (TJ * KDIM) / 1024; ++r) {
                int o = t4 + r * 1024;
                async_ld_b128(Aj + j0 * KDIM + o, sA + o);
                async_ld_b128(Bj + j0 * KDIM + o, sB + o);
                async_ld_b128(Cj + j0 * KDIM + o, sC + o);
            }
            wait_async0();
        }
#else
        for (int t = threadIdx.x; t < TJ * KDIM; t += blockDim.x) {
            sA[t] = Aj[j0 * KDIM + t];
            sB[t] = Bj[j0 * KDIM + t];
            sC[t] = Cj[j0 * KDIM + t];
        }
#endif
        __syncthreads();
#pragma unroll 4
        for (int jj = 0; jj < TJ; ++jj) {
            float x  = fmaf(sA[jj * KDIM + lane], z2,
                       fmaf(sB[jj * KDIM + lane], z1, sC[jj * KDIM + lane]));
            float nm = fmaxf(m, x);
            s = fmaf(s, __expf(m - nm), __expf(x - nm));
            m = nm;
        }
    }
    float lqk = m + __logf(s) - log_norm;   // log q_k(z_ik)
    float tot = wave_sum(lqk);
    if (lane == 0) lprod[i] = tot;          // log prod_k q_k(z_ik)
}

// ---------------------------------------------------------------------------
// Kernel 3: S[i,j] = Csum[j] + X[i,:]·W[j,:]  via V_WMMA_F32_16X16X4_F32,
// fused with online logsumexp over j per row.  One full wave per 16-row strip
// (EXEC all ones, uniform control flow around all WMMAs).
// A 16x4 f32 layout: lanes 0-15 = M rows; VGPR0 holds K={0(lo half),2(hi half)},
// VGPR1 holds K={1,3}.  B mirrors with N on lanes.  C/D: VGPR r = rows r / r+8,
// lanes = columns.
// ---------------------------------------------------------------------------
__global__ void k_wmma_lse(const float* __restrict__ X, const float* __restrict__ Wp,
                           const float* __restrict__ Csum,
                           float* __restrict__ lqz, float log_norm, int I) {
    int lane = threadIdx.x;       // 32 threads = 1 wave
    int i0   = blockIdx.x * 16;
    int half = lane >> 4;
    int l15  = lane & 15;

    v2f afrag[16];
    int arow = i0 + l15;
#pragma unroll
    for (int s = 0; s < 16; ++s) {
        afrag[s] = *(const v2f*)(X + arow * FDIM + s * 4 + half * 2);
    }

    float m[8], sm[8];
#pragma unroll
    for (int r = 0; r < 8; ++r) { m[r] = -INFINITY; sm[r] = 0.f; }

    for (int j0 = 0; j0 < I; j0 += 16) {
        int col = j0 + l15;
        v8f c = {};
#pragma unroll
        for (int s = 0; s < 16; ++s) {
            int pairIdx = s * 2 + half;   // = (s*4 + half*2) / 2
            v2f b = *(const v2f*)(Wp + (size_t)pairIdx * (2 * I) + 2 * col);
            c = __builtin_amdgcn_wmma_f32_16x16x4_f32(
                    false, afrag[s], false, b, (short)0, c, false, false);
        }
        float csj = Csum[col];
#pragma unroll
        for (int r = 0; r < 8; ++r) {
            float x  = c[r] + csj;
            float nm = fmaxf(m[r], x);
            sm[r] = fmaf(sm[r], __expf(m[r] - nm), __expf(x - nm));
            m[r]  = nm;
        }
    }

    // merge (m,s) across the 16 lanes holding each row's columns
#pragma unroll
    for (int r = 0; r < 8; ++r) {
#pragma unroll
        for (int mask = 1; mask < 16; mask <<= 1) {
            float om = __shfl_xor(m[r], mask, 32);
            float os = __shfl_xor(sm[r], mask, 32);
            float nm = fmaxf(m[r], om);
            sm[r] = sm[r] * __expf(m[r] - nm) + os * __expf(om - nm);
            m[r]  = nm;
        }
        if (l15 == 0) {
            int row = r + 8 * half;
            lqz[i0 + row] = m[r] + __logf(sm[r]) - log_norm;   // log q(z_i)
        }
    }
}

// ---------------------------------------------------------------------------
// Kernel 4: stable BCE mean per sample: bce = softplus(-x) + (1-p)*x
// ---------------------------------------------------------------------------
__global__ void k_bce(const float* __restrict__ x_logits, const float* __restrict__ x_probs,
                      float* __restrict__ rec, int PIX) {
    int i = blockIdx.x;
    const float* xr = x_logits + (size_t)i * PIX;
    const float* pr = x_probs  + (size_t)i * PIX;
    float acc = 0.f;
    for (int t = threadIdx.x; t < PIX; t += blockDim.x) {
        float x = xr[t], p = pr[t];
        float spm = (x > 0.f) ? log1pf(__expf(-x)) : (-x + log1pf(__expf(x)));
        acc += spm + (1.f - p) * x;
    }
    __shared__ float part[8];
    float w = wave_sum(acc);
    int wid = threadIdx.x >> 5, lane = threadIdx.x & 31;
    if (lane == 0) part[wid] = w;
    __syncthreads();
    if (wid == 0) {
        float v = (lane < 8) ? part[lane] : 0.f;
        v = wave_sum(v);
        if (lane == 0) rec[i] = v / (float)PIX;
    }
}

// ---------------------------------------------------------------------------
// Kernel 5: per-sample totals (faithful TC+MI+KL terms) and deterministic mean
// ---------------------------------------------------------------------------
__global__ void k_final(const float* __restrict__ rec,
                        const float* __restrict__ lqx, const float* __restrict__ lpz,
                        const float* __restrict__ lqz, const float* __restrict__ lprod,
                        float* __restrict__ out, int I) {
    const float inv_xdim = 1.0f / 12288.0f;   // X_DIM = 64*64*3
    float acc = 0.f;
    for (int i = threadIdx.x; i < I; i += blockDim.x) {
        float tc = (lqz[i]  - lprod[i]) * inv_xdim;
        float mi = (lqx[i]  - lqz[i])   * inv_xdim;
        float dk = (lprod[i] - lpz[i])  * inv_xdim;
        acc += rec[i] + tc + mi + dk;
    }
    __shared__ float part[8];
    float w = wave_sum(acc);
    int wid = threadIdx.x >> 5, lane = threadIdx.x & 31;
    if (lane == 0) part[wid] = w;
    __syncthreads();
    if (wid == 0) {
        float v = (lane < 8) ? part[lane] : 0.f;
        v = wave_sum(v);
        if (lane == 0) out[0] = v / (float)I;
    }
}

extern "C" void kernel_launch(void* const* d_in, const int* in_sizes, int n_in,
                              void* d_out, int out_size, void* d_ws, size_t ws_size,
                              hipStream_t stream) {
    const float* mu_q    = (const float*)d_in[0];
    const float* rho_q   = (const float*)d_in[1];
    const float* eps     = (const float*)d_in[2];
    const float* x_logit = (const float*)d_in[3];
    const float* x_probs = (const float*)d_in[4];
    float* out = (float*)d_out;

    int I   = in_sizes[0] / KDIM;        // 2048
    int PIX = in_sizes[3] / I;           // 12288
    float log_norm = logf((float)I) + logf(100000.0f);

    float* p    = (float*)d_ws;
    float* X    = p; p += (size_t)I * FDIM;
    float* Wp   = p; p += (size_t)FDIM * I;
    float* Aj   = p; p += (size_t)I * KDIM;
    float* Bj   = p; p += (size_t)I * KDIM;
    float* Cj   = p; p += (size_t)I * KDIM;
    float* Csum = p; p += I;
    float* lqx  = p; p += I;
    float* lpz  = p; p += I;
    float* lprod= p; p += I;
    float* lqz  = p; p += I;
    float* rec  = p; p += I;

    k_prep    <<<I,      32,  0, stream>>>(mu_q, rho_q, eps, X, Wp, Aj, Bj, Cj, Csum, lqx, lpz, I);
    k_perk_lse<<<I / 8,  256, 0, stream>>>(X, Aj, Bj, Cj, lprod, log_norm, I);
    k_wmma_lse<<<I / 16, 32,  0, stream>>>(X, Wp, Csum, lqz, log_norm, I);
    k_bce     <<<I,      256, 0, stream>>>(x_logit, x_probs, rec, PIX);
    k_final   <<<1,      256, 0, stream>>>(rec, lqx, lpz, lqz, lprod, out, I);
}